// Net_20160576487463
// MI455X (gfx1250) — compile-verified
//
#include <hip/hip_runtime.h>

// CDNA5 / gfx1250 (MI455X): wave32, WMMA 16x16x4 f32, NT stores for the
// 256MB output stream, RT stores for the L2-resident 16MB score tables.

typedef __attribute__((ext_vector_type(2))) float v2f;
typedef __attribute__((ext_vector_type(8))) float v8f;

#define HDIM 10
#define FDIM 20

// ---------------------------------------------------------------------------
// Phase 1: per-node attention scores.
//   e_l[n,k] = sum_h Z[n,h,k] * a_l[h,k]   (and e_r with a_r)
// Thread per (n,k). Consecutive threads -> consecutive addresses at each h
// step (coalesced). a_l/a_r (800B each) staged in LDS once per block.
// Stores use default (RT) temporal hint so e_l/e_r stay resident in the
// 192MB L2 for the phase-2 gathers.
// ---------------------------------------------------------------------------
__global__ void node_scores_kernel(const float* __restrict__ Z,
                                   const float* __restrict__ a_l,
                                   const float* __restrict__ a_r,
                                   float* __restrict__ e_l,
                                   float* __restrict__ e_r,
                                   int n_nodes) {
  __shared__ float sal[HDIM * FDIM];
  __shared__ float sar[HDIM * FDIM];
  if (threadIdx.x < HDIM * FDIM) {
    sal[threadIdx.x] = a_l[threadIdx.x];
    sar[threadIdx.x] = a_r[threadIdx.x];
  }
  __syncthreads();

  int t = blockIdx.x * blockDim.x + threadIdx.x;
  int total = n_nodes * FDIM;
  if (t >= total) return;
  int n = t / FDIM;
  int k = t - n * FDIM;

  const float* z = Z + (size_t)n * (HDIM * FDIM) + k;
  float sl = 0.0f, sr = 0.0f;
#pragma unroll
  for (int h = 0; h < HDIM; ++h) {
    float zv = z[h * FDIM];
    sl = __builtin_fmaf(zv, sal[h * FDIM + k], sl);
    sr = __builtin_fmaf(zv, sar[h * FDIM + k], sr);
  }
  e_l[t] = sl;
  e_r[t] = sr;
}

// ---------------------------------------------------------------------------
// Phase 2: out[e,:] = e_l[row[e],:] + e_r[col[e],:]
// One wave32 handles a tile of 16 edges. Feature columns 0..15 are computed
// with 4 chained V_WMMA_F32_16X16X4_F32:
//   D = sum_s A_s * B_s + C,   B_s[j,k] = delta(k, 4s+j)  (shifted identity)
//   A_s = gathered e_l tile, columns 4s..4s+3 ; C = gathered e_r tile.
// VGPR layouts per CDNA5 ISA 7.12.2:
//   A (16x4 f32): lane L -> rows m=L&15, K = 2*(L>>4) (+1 in 2nd VGPR)
//   C/D (16x16 f32): lane L, VGPR v -> element [v + 8*(L>>4), L&15]
// Columns 16..19 are done with a scalar float2 path.
// EXEC is all-ones around the WMMAs: the tile guard is wave-uniform and the
// edge count is a multiple of 16 (tail edges go to a separate scalar kernel).
// ---------------------------------------------------------------------------
__global__ void edge_gather_wmma_kernel(const int* __restrict__ row,
                                        const int* __restrict__ col,
                                        const float* __restrict__ e_l,
                                        const float* __restrict__ e_r,
                                        float* __restrict__ out,
                                        int n_tiles) {
  const int lane = threadIdx.x & 31;
  const int wave = (blockIdx.x * blockDim.x + threadIdx.x) >> 5;
  if (wave >= n_tiles) return;  // wave-uniform: EXEC stays all-ones

  const int eb = wave * 16;
  const int m = lane & 15;      // tile row (edge within tile) / N column
  const int half = lane >> 4;   // 0: lanes 0-15, 1: lanes 16-31
  const int kk = half * 2;      // K-pair owned by this lane half (A/B layout)

  // Prefetch index data a few tiles ahead (global_prefetch_b8).
  __builtin_prefetch(row + eb + 256, 0, 1);
  __builtin_prefetch(col + eb + 256, 0, 1);

  const int r_own = row[eb + m];
  const int c_own = col[eb + m];

  // ---- C accumulator = gathered e_r tile (C/D striping) ------------------
  // For fixed v, lanes of one half share the col index -> each load is a
  // 64B-contiguous segment per half (L2-friendly).
  v8f acc;
#pragma unroll
  for (int v = 0; v < 8; ++v) {
    int ci = col[eb + v + 8 * half];
    acc[v] = e_r[(size_t)ci * FDIM + m];
  }

  // ---- 4 chained WMMAs: add e_l columns 0..15 ----------------------------
  const float* elr = e_l + (size_t)r_own * FDIM;  // 16B-aligned (20 floats)
#pragma unroll
  for (int s = 0; s < 4; ++s) {
    v2f a = *(const v2f*)(elr + 4 * s + kk);  // A[m, 4s+kk], A[m, 4s+kk+1]
    v2f b;
    b.x = (m == 4 * s + kk) ? 1.0f : 0.0f;        // B[4s+kk,   m]
    b.y = (m == 4 * s + kk + 1) ? 1.0f : 0.0f;    // B[4s+kk+1, m]
    acc = __builtin_amdgcn_wmma_f32_16x16x4_f32(
        /*neg_a=*/false, a, /*neg_b=*/false, b,
        /*c_mod=*/(short)0, acc, /*reuse_a=*/false, /*reuse_b=*/false);
  }

  // ---- store D (columns 0..15), non-temporal: 256MB stream must not evict
  // the L2-resident score tables. Per v: two 64B coalesced segments.
#pragma unroll
  for (int v = 0; v < 8; ++v) {
    size_t o = (size_t)(eb + v + 8 * half) * FDIM + m;
    __builtin_nontemporal_store(acc[v], out + o);
  }

  // ---- columns 16..19: lane handles edge m, cols 16+kk and 17+kk ---------
  {
    const int cb = 16 + kk;
    const float* err = e_r + (size_t)c_own * FDIM;
    float r0 = elr[cb] + err[cb];
    float r1 = elr[cb + 1] + err[cb + 1];
    size_t o = (size_t)(eb + m) * FDIM + cb;
    __builtin_nontemporal_store(r0, out + o);
    __builtin_nontemporal_store(r1, out + o + 1);
  }
}

// Scalar fallback for a (<16)-edge tail. Not launched for E % 16 == 0.
__global__ void edge_tail_kernel(const int* __restrict__ row,
                                 const int* __restrict__ col,
                                 const float* __restrict__ e_l,
                                 const float* __restrict__ e_r,
                                 float* __restrict__ out,
                                 int e0, int n_edges) {
  int t = blockIdx.x * blockDim.x + threadIdx.x;
  int e = e0 + t / FDIM;
  int k = t - (t / FDIM) * FDIM;
  if (e >= n_edges) return;
  float v = e_l[(size_t)row[e] * FDIM + k] + e_r[(size_t)col[e] * FDIM + k];
  __builtin_nontemporal_store(v, out + (size_t)e * FDIM + k);
}

extern "C" void kernel_launch(void* const* d_in, const int* in_sizes, int n_in,
                              void* d_out, int out_size, void* d_ws, size_t ws_size,
                              hipStream_t stream) {
  (void)n_in; (void)out_size; (void)ws_size;
  const float* Z   = (const float*)d_in[0];
  const int*   row = (const int*)d_in[1];
  const int*   col = (const int*)d_in[2];
  const float* a_l = (const float*)d_in[3];
  const float* a_r = (const float*)d_in[4];
  float* out = (float*)d_out;

  const int n_nodes = in_sizes[0] / (HDIM * FDIM);  // 100000
  const int n_edges = in_sizes[1];                  // 3200000

  // Workspace: e_l then e_r, n_nodes*FDIM floats each (16 MB total).
  float* e_l = (float*)d_ws;
  float* e_r = e_l + (size_t)n_nodes * FDIM;

  {
    int total = n_nodes * FDIM;
    int blocks = (total + 255) / 256;
    node_scores_kernel<<<blocks, 256, 0, stream>>>(Z, a_l, a_r, e_l, e_r, n_nodes);
  }
  {
    int n_tiles = n_edges / 16;                 // 200000 full tiles
    int waves_per_block = 256 / 32;             // wave32
    int blocks = (n_tiles + waves_per_block - 1) / waves_per_block;
    edge_gather_wmma_kernel<<<blocks, 256, 0, stream>>>(row, col, e_l, e_r, out, n_tiles);

    int tail = n_edges - n_tiles * 16;
    if (tail > 0) {
      int tt = tail * FDIM;
      edge_tail_kernel<<<(tt + 255) / 256, 256, 0, stream>>>(row, col, e_l, e_r,
                                                             out, n_tiles * 16, n_edges);
    }
  }
}